// ChamferLoss_2877628088764
// MI455X (gfx1250) — compile-verified
//
#include <hip/hip_runtime.h>

typedef __attribute__((ext_vector_type(2))) float v2f;
typedef __attribute__((ext_vector_type(8))) float v8f;

#define WAVES 8
#define BIGF  3.0e38f

// One 16x16x4 f32 WMMA tile step: given A operand (rows, premultiplied by -2,
// pad column = 1) compute D[r][c] = yy_c - 2*x_r.y_c for a 16-column tile and
// fold into the running per-element min accumulators. The min is done with raw
// v_min_num_f32 (same HW op fminf lowers to) to avoid the compiler inserting
// canonicalize (v_max_num x,x,x) ops on both operands every iteration.
__device__ __forceinline__ void tile_step(const float* __restrict__ yb, int c,
                                          int half, v2f a, const v8f& zc,
                                          float m[8]) {
  const float* yc = yb + (size_t)c * 3;
  float y0 = yc[0], y1 = yc[1], y2 = yc[2];
  float yy = y0 * y0 + y1 * y1 + y2 * y2;
  v2f bb;
  bb.x = half ? y2 : y0;   // K=0 (lanes 0-15) / K=2 (lanes 16-31)
  bb.y = half ? yy : y1;   // K=1 (lanes 0-15) / K=3 (lanes 16-31)
  v8f d = __builtin_amdgcn_wmma_f32_16x16x4_f32(
      /*neg_a=*/false, a, /*neg_b=*/false, bb,
      /*c_mod=*/(short)0, zc, /*reuse_a=*/false, /*reuse_b=*/false);
#pragma unroll
  for (int v = 0; v < 8; ++v) {
    float dv = d[v];
    asm("v_min_num_f32 %0, %0, %1" : "+v"(m[v]) : "v"(dv));
  }
}

// X: [B, R, 3] (rows of the distance tile), Y: [B, C, 3] (columns).
// outmin[b*R + r] = ||x_r||^2 + min_c ( ||y_c||^2 - 2 x_r . y_c )
__global__ __launch_bounds__(256) void chamfer_min_pass(
    const float* __restrict__ X, const float* __restrict__ Y,
    float* __restrict__ outmin, int R, int C) {
  __shared__ float lds[WAVES * 16];

  const int tid  = threadIdx.x;
  const int wave = tid >> 5;
  const int lane = tid & 31;
  const int l16  = lane & 15;
  const int half = lane >> 4;
  const int b    = blockIdx.y;
  const int r0   = blockIdx.x * 16;

  const float* xb = X + ((size_t)b * R + r0) * 3;
  const float* yb = Y + (size_t)b * C * 3;

  // Build the A operand (held for the whole kernel).
  float x0 = xb[l16 * 3 + 0];
  float x1 = xb[l16 * 3 + 1];
  float x2 = xb[l16 * 3 + 2];
  v2f a;
  a.x = half ? (-2.0f * x2) : (-2.0f * x0);  // K=0 / K=2
  a.y = half ? 1.0f         : (-2.0f * x1);  // K=1 / K=3 (pad K: 1 * yy_c)

  v8f zc = {0.f, 0.f, 0.f, 0.f, 0.f, 0.f, 0.f, 0.f};
  float m[8];
#pragma unroll
  for (int v = 0; v < 8; ++v) m[v] = BIGF;

  const int CT = C >> 4;  // number of 16-column tiles
  for (int mt = wave * 2; mt < CT; mt += WAVES * 2) {
    // software prefetch of a future tile (speculative; OOB is dropped)
    __builtin_prefetch(yb + (size_t)(((mt + WAVES * 2) << 4) + l16) * 3, 0, 3);
    tile_step(yb, (mt << 4) + l16, half, a, zc, m);
    if (mt + 1 < CT) tile_step(yb, ((mt + 1) << 4) + l16, half, a, zc, m);
  }

  // Min across the 16 columns of this wave's tiles: D row v+8*half lives in
  // accumulator v across a 16-lane group -> xor-shuffle tree within width 16.
#pragma unroll
  for (int v = 0; v < 8; ++v) {
#pragma unroll
    for (int off = 8; off >= 1; off >>= 1)
      m[v] = fminf(m[v], __shfl_xor(m[v], off, 16));
  }

  // One lane per (local row) publishes its wave-partial min to LDS.
#pragma unroll
  for (int v = 0; v < 8; ++v)
    if (l16 == v) lds[wave * 16 + half * 8 + v] = m[v];
  __syncthreads();

  // Combine the 8 waves' partials and add ||x_r||^2.
  if (tid < 16) {
    float r = lds[tid];
#pragma unroll
    for (int w = 1; w < WAVES; ++w) r = fminf(r, lds[w * 16 + tid]);
    int row = r0 + tid;
    const float* xr = X + ((size_t)b * R + row) * 3;
    float xx = xr[0] * xr[0] + xr[1] * xr[1] + xr[2] * xr[2];
    outmin[(size_t)b * R + row] = xx + r;
  }
}

// Deterministic single-block reduction: out = mean(rowmin) + mean(colmin).
__global__ __launch_bounds__(1024) void chamfer_reduce(
    const float* __restrict__ bufA, int nA, float invA,
    const float* __restrict__ bufB, int nB, float invB,
    float* __restrict__ out) {
  __shared__ float s[1024];
  float a1 = 0.f, a2 = 0.f;
  for (int i = threadIdx.x; i < nA; i += 1024) a1 += bufA[i];
  for (int i = threadIdx.x; i < nB; i += 1024) a2 += bufB[i];
  s[threadIdx.x] = a1 * invA + a2 * invB;
  __syncthreads();
  for (int off = 512; off > 0; off >>= 1) {
    if (threadIdx.x < off) s[threadIdx.x] += s[threadIdx.x + off];
    __syncthreads();
  }
  if (threadIdx.x == 0) out[0] = s[0];
}

extern "C" void kernel_launch(void* const* d_in, const int* in_sizes, int n_in,
                              void* d_out, int out_size, void* d_ws,
                              size_t ws_size, hipStream_t stream) {
  const float* preds = (const float*)d_in[0];  // [B, M, 3]
  const float* gts   = (const float*)d_in[1];  // [B, N, 3]
  const int B = 8, D = 3;
  const int M = in_sizes[0] / (B * D);
  const int N = in_sizes[1] / (B * D);

  float* rowmin = (float*)d_ws;             // [B*N]: per-gt min over preds (loss_2)
  float* colmin = rowmin + (size_t)B * N;   // [B*M]: per-pred min over gts (loss_1)

  dim3 blk(256);
  chamfer_min_pass<<<dim3(N / 16, B), blk, 0, stream>>>(gts, preds, rowmin, N, M);
  chamfer_min_pass<<<dim3(M / 16, B), blk, 0, stream>>>(preds, gts, colmin, M, N);
  chamfer_reduce<<<1, 1024, 0, stream>>>(rowmin, B * N, 1.0f / (float)(B * N),
                                         colmin, B * M, 1.0f / (float)(B * M),
                                         (float*)d_out);
}